// ST_GNN_53171695124552
// MI455X (gfx1250) — compile-verified
//
#include <hip/hip_runtime.h>

#define NN 50000
#define EE 500000
#define EF (EE + NN)
#define HIDC 128

typedef __attribute__((ext_vector_type(2))) float v2f;
typedef __attribute__((ext_vector_type(8))) float v8f;

__device__ __forceinline__ v8f wmma4(v2f a, v2f b, v8f c) {
  // D = A(16x4 f32) * B(4x16 f32) + C(16x16 f32)
  return __builtin_amdgcn_wmma_f32_16x16x4_f32(false, a, false, b, (short)0, c, false, false);
}

// ---------------------------------------------------------------------------
// Mean of edge_attr over edges -> ea_sum[32] (scaled later)
__global__ void ea_reduce_kernel(const float* __restrict__ ea, float* __restrict__ ea_sum) {
  int tid = blockIdx.x * blockDim.x + threadIdx.x;   // 64*256 = 16384 threads
  int c = tid & 31;
  int r0 = tid >> 5;                                  // 512 row-threads per column
  float acc = 0.f;
  for (int r = r0; r < EE; r += 512) acc += ea[(size_t)r * 32 + c];
  atomicAdd(&ea_sum[c], acc);
}

// self-loop edge features: sl1[c] = mean_attr @ We1, sl2[c] = mean_attr @ We2
__global__ void sl_ea_kernel(const float* __restrict__ ea_sum,
                             const float* __restrict__ We1, const float* __restrict__ We2,
                             float* __restrict__ sl1, float* __restrict__ sl2) {
  int c = threadIdx.x;  // 384 threads
  const float inv = 1.f / (float)EE;
  if (c < 256) {
    float a = 0.f;
#pragma unroll
    for (int k = 0; k < 32; ++k) a += (ea_sum[k] * inv) * We1[k * 256 + c];
    sl1[c] = a;
  } else {
    int cc = c - 256;
    float a = 0.f;
#pragma unroll
    for (int k = 0; k < 32; ++k) a += (ea_sum[k] * inv) * We2[k * 128 + cc];
    sl2[cc] = a;
  }
}

// ---------------------------------------------------------------------------
// Node GEMM: C[M x N] = A[M x 128] @ B + bias.  4 waves/block, each a 16x16 tile.
// BTRANS==false: B row-major [128 x LDB];  BTRANS==true: B is W[N x LDB], use W[n][k].
// All strides compile-time so the K-loop is branch-free straight-line code.
template <int LDB, int LDC, bool BTRANS>
__global__ void __launch_bounds__(128)
node_gemm_wmma(const float* __restrict__ A, const float* __restrict__ B,
               const float* __restrict__ bias, float* __restrict__ C) {
  __shared__ float As[16 * 132];
  const int tid = threadIdx.x;
  const int row0 = blockIdx.x * 16;
#pragma unroll
  for (int i = 0; i < 16; ++i) {
    int idx = tid + i * 128;
    int r = idx >> 7, c = idx & 127;
    As[r * 132 + c] = A[(size_t)(row0 + r) * 128 + c];
  }
  __syncthreads();
  const int w = tid >> 5, lane = tid & 31;
  const int m16 = lane & 15, half = lane >> 4;
  const int col0 = (blockIdx.y * 4 + w) * 16;
  const float* bp = BTRANS ? (B + (size_t)(col0 + m16) * LDB)
                           : (B + col0 + m16);
  const float* ap = As + m16 * 132 + 2 * half;
  v8f acc = {};
#pragma unroll
  for (int k = 0; k < 128; k += 4) {
    const int kb = k + 2 * half;
    v2f a, b;
    a.x = ap[k];
    a.y = ap[k + 1];
    if (BTRANS) {
      b.x = bp[kb];
      b.y = bp[kb + 1];
    } else {
      b.x = bp[(size_t)kb * LDB];
      b.y = bp[(size_t)(kb + 1) * LDB];
    }
    acc = wmma4(a, b, acc);
  }
#pragma unroll
  for (int j = 0; j < 8; ++j) {
    int r = row0 + j + 8 * half;
    int cc = col0 + m16;
    C[(size_t)r * LDC + cc] = acc[j] + bias[cc];
  }
}

// ---------------------------------------------------------------------------
// GATv2 pass A: per-(edge,head) wave computes attention logit; atomicMax lmax.
template <int H>
__global__ void __launch_bounds__(256)
gat_logit_kernel(const float* __restrict__ xl, const float* __restrict__ xr,
                 const int* __restrict__ src, const int* __restrict__ dst,
                 const float* __restrict__ edge_attr, const float* __restrict__ sl_ea,
                 const float* __restrict__ We, const float* __restrict__ att,
                 float* __restrict__ logit, unsigned* __restrict__ lmax) {
  const int gid = blockIdx.x * blockDim.x + threadIdx.x;
  const int wid = gid >> 5, lane = gid & 31;
  if (wid >= EF * H) return;
  const int e = (H == 1) ? wid : (wid >> 1);
  const int hh = (H == 1) ? 0 : (wid & 1);
  const int HC = H * HIDC;
  int s, d;
  float eav[4];
  if (e >= EE) {
    s = e - EE; d = s;
#pragma unroll
    for (int j = 0; j < 4; ++j) eav[j] = sl_ea[hh * HIDC + lane + 32 * j];
  } else {
    s = src[e]; d = dst[e];
#pragma unroll
    for (int j = 0; j < 4; ++j) eav[j] = 0.f;
    const float* ear = edge_attr + (size_t)e * 32;
#pragma unroll 4
    for (int k = 0; k < 32; ++k) {
      float ek = ear[k];
      const float* wr = We + (size_t)k * HC + hh * HIDC;
#pragma unroll
      for (int j = 0; j < 4; ++j) eav[j] += ek * wr[lane + 32 * j];
    }
  }
  const float* xlr = xl + (size_t)s * HC + hh * HIDC;
  const float* xrr = xr + (size_t)d * HC + hh * HIDC;
  const float* atr = att + hh * HIDC;
  float accv = 0.f;
#pragma unroll
  for (int j = 0; j < 4; ++j) {
    int c = lane + 32 * j;
    float m = xlr[c] + xrr[c] + eav[j];
    m = (m > 0.f) ? m : 0.2f * m;     // leaky_relu 0.2
    accv += m * atr[c];
  }
#pragma unroll
  for (int off = 16; off >= 1; off >>= 1) accv += __shfl_xor(accv, off, 32);
  if (lane == 0) {
    logit[(size_t)e * H + hh] = accv;
    unsigned bits = __float_as_uint(accv);
    unsigned key = (bits & 0x80000000u) ? ~bits : (bits | 0x80000000u);
    atomicMax(&lmax[(size_t)d * H + hh], key);
  }
}

// pass B: p = exp(logit - lmax[dst]); denom[dst] += p (p overwrites logit)
template <int H>
__global__ void gat_softmax_kernel(float* __restrict__ logit, const unsigned* __restrict__ lmax,
                                   float* __restrict__ denom, const int* __restrict__ dst) {
  int idx = blockIdx.x * blockDim.x + threadIdx.x;
  if (idx >= EF * H) return;
  int e = (H == 1) ? idx : (idx >> 1);
  int hh = (H == 1) ? 0 : (idx & 1);
  int d = (e >= EE) ? (e - EE) : dst[e];
  unsigned key = lmax[(size_t)d * H + hh];
  unsigned bits = (key & 0x80000000u) ? (key & 0x7FFFFFFFu) : ~key;
  float p = expf(logit[idx] - __uint_as_float(bits));
  logit[idx] = p;
  atomicAdd(&denom[(size_t)d * H + hh], p);
}

// pass C: acc[dst] += (p/denom[dst]) * xl[src]
template <int H>
__global__ void __launch_bounds__(256)
gat_aggregate_kernel(const float* __restrict__ xl, const float* __restrict__ p,
                     const float* __restrict__ denom, const int* __restrict__ src,
                     const int* __restrict__ dst, float* __restrict__ acc) {
  const int gid = blockIdx.x * blockDim.x + threadIdx.x;
  const int wid = gid >> 5, lane = gid & 31;
  if (wid >= EF * H) return;
  const int e = (H == 1) ? wid : (wid >> 1);
  const int hh = (H == 1) ? 0 : (wid & 1);
  int s, d;
  if (e >= EE) { s = e - EE; d = s; } else { s = src[e]; d = dst[e]; }
  const int HC = H * HIDC;
  float alpha = p[(size_t)e * H + hh] / denom[(size_t)d * H + hh];
  const float* xlr = xl + (size_t)s * HC + hh * HIDC;
  float* ar = acc + (size_t)d * HC + hh * HIDC;
#pragma unroll
  for (int j = 0; j < 4; ++j) {
    int c = lane + 32 * j;
    atomicAdd(&ar[c], alpha * xlr[c]);
  }
}

// z = elu(mean_h(acc) + bias)
template <int H>
__global__ void finalize_elu_kernel(const float* __restrict__ acc, const float* __restrict__ bias,
                                    float* __restrict__ out) {
  int idx = blockIdx.x * blockDim.x + threadIdx.x;
  if (idx >= NN * HIDC) return;
  int n = idx >> 7, c = idx & 127;
  float v;
  if (H == 2) v = 0.5f * (acc[(size_t)n * 256 + c] + acc[(size_t)n * 256 + 128 + c]);
  else        v = acc[(size_t)n * 128 + c];
  v += bias[c];
  out[idx] = (v > 0.f) ? v : (expf(v) - 1.f);
}

// GRU with h_prev = 0: gh = b_hh; h = (1-sigmoid(iz+hz))*tanh(ig + sigmoid(ir+hr)*hg)
__global__ void gru_kernel(const float* __restrict__ gi, const float* __restrict__ b_hh,
                           float* __restrict__ h) {
  int idx = blockIdx.x * blockDim.x + threadIdx.x;
  if (idx >= NN * HIDC) return;
  int n = idx >> 7, c = idx & 127;
  const float* g = gi + (size_t)n * 384;
  float r = 1.f / (1.f + expf(-(g[c] + b_hh[c])));
  float u = 1.f / (1.f + expf(-(g[128 + c] + b_hh[128 + c])));
  float gg = tanhf(g[256 + c] + r * b_hh[256 + c]);
  h[idx] = (1.f - u) * gg;
}

// ---------------------------------------------------------------------------
// Fused edge classifier: rep=[h[src],h[dst],ea] (16x288 in LDS) -> WMMA Wc1 ->
// relu -> WMMA Wc2 -> relu -> Wc3 -> out[16x2]. 8 waves, E/16 blocks.
__global__ void __launch_bounds__(256)
edge_classifier_kernel(const float* __restrict__ h, const int* __restrict__ src,
                       const int* __restrict__ dst, const float* __restrict__ edge_attr,
                       const float* __restrict__ Wc1, const float* __restrict__ bc1,
                       const float* __restrict__ Wc2, const float* __restrict__ bc2,
                       const float* __restrict__ Wc3, const float* __restrict__ bc3,
                       float* __restrict__ out) {
  __shared__ float rep[16 * 292];
  __shared__ float h1s[16 * 132];
  __shared__ float h2s[16 * 68];
  const int tid = threadIdx.x;
  const int e0 = blockIdx.x * 16;
#pragma unroll
  for (int i = 0; i < 18; ++i) {
    int idx = tid + i * 256;
    int r = idx / 288, c = idx - r * 288;
    int e = e0 + r;
    float v;
    if (c < 128)      v = h[(size_t)src[e] * 128 + c];
    else if (c < 256) v = h[(size_t)dst[e] * 128 + (c - 128)];
    else              v = edge_attr[(size_t)e * 32 + (c - 256)];
    rep[r * 292 + c] = v;
  }
  __syncthreads();
  const int w = tid >> 5, lane = tid & 31;
  const int m16 = lane & 15, half = lane >> 4;
  {  // stage 1: h1 = relu(rep @ Wc1 + bc1), N=128, all 8 waves
    const int col0 = w * 16;
    const float* ap = rep + m16 * 292 + 2 * half;
    const float* bp = Wc1 + col0 + m16;
    v8f acc = {};
#pragma unroll 8
    for (int k = 0; k < 288; k += 4) {
      const int kb = k + 2 * half;
      v2f a, b;
      a.x = ap[k];
      a.y = ap[k + 1];
      b.x = bp[(size_t)kb * 128];
      b.y = bp[(size_t)(kb + 1) * 128];
      acc = wmma4(a, b, acc);
    }
#pragma unroll
    for (int j = 0; j < 8; ++j) {
      int r = j + 8 * half, cc = col0 + m16;
      float v = acc[j] + bc1[cc];
      h1s[r * 132 + cc] = fmaxf(v, 0.f);
    }
  }
  __syncthreads();
  if (w < 4) {  // stage 2: h2 = relu(h1 @ Wc2 + bc2), N=64
    const int col0 = w * 16;
    const float* ap = h1s + m16 * 132 + 2 * half;
    const float* bp = Wc2 + col0 + m16;
    v8f acc = {};
#pragma unroll
    for (int k = 0; k < 128; k += 4) {
      const int kb = k + 2 * half;
      v2f a, b;
      a.x = ap[k];
      a.y = ap[k + 1];
      b.x = bp[(size_t)kb * 64];
      b.y = bp[(size_t)(kb + 1) * 64];
      acc = wmma4(a, b, acc);
    }
#pragma unroll
    for (int j = 0; j < 8; ++j) {
      int r = j + 8 * half, cc = col0 + m16;
      float v = acc[j] + bc2[cc];
      h2s[r * 68 + cc] = fmaxf(v, 0.f);
    }
  }
  __syncthreads();
  if (tid < 32) {  // stage 3: out = h2 @ Wc3 + bc3 (N=2)
    int r = tid >> 1, o = tid & 1;
    float v = bc3[o];
#pragma unroll 8
    for (int k = 0; k < 64; ++k) v += h2s[r * 68 + k] * Wc3[k * 2 + o];
    out[(size_t)(e0 + r) * 2 + o] = v;
  }
}

// ---------------------------------------------------------------------------
extern "C" void kernel_launch(void* const* d_in, const int* in_sizes, int n_in,
                              void* d_out, int out_size, void* d_ws, size_t ws_size,
                              hipStream_t stream) {
  const float* x    = (const float*)d_in[0];
  const int*   ei   = (const int*)d_in[1];
  const float* ea   = (const float*)d_in[2];
  const float* Wl1  = (const float*)d_in[4];
  const float* bl1  = (const float*)d_in[5];
  const float* Wr1  = (const float*)d_in[6];
  const float* br1  = (const float*)d_in[7];
  const float* We1  = (const float*)d_in[8];
  const float* att1 = (const float*)d_in[9];
  const float* bias1= (const float*)d_in[10];
  const float* Wl2  = (const float*)d_in[11];
  const float* bl2  = (const float*)d_in[12];
  const float* Wr2  = (const float*)d_in[13];
  const float* br2  = (const float*)d_in[14];
  const float* We2  = (const float*)d_in[15];
  const float* att2 = (const float*)d_in[16];
  const float* bias2= (const float*)d_in[17];
  const float* W_ih = (const float*)d_in[18];
  const float* b_ih = (const float*)d_in[20];
  const float* b_hh = (const float*)d_in[21];
  const float* Wc1  = (const float*)d_in[22];
  const float* bc1  = (const float*)d_in[23];
  const float* Wc2  = (const float*)d_in[24];
  const float* bc2  = (const float*)d_in[25];
  const float* Wc3  = (const float*)d_in[26];
  const float* bc3  = (const float*)d_in[27];
  float* outp = (float*)d_out;

  const int* srcp = ei;
  const int* dstp = ei + EE;

  // workspace arena (floats)
  float* ws = (float*)d_ws;
  const size_t n = NN;
  float*    R1    = ws;                 // N*384: xl1, later gi
  float*    R2    = R1 + n * 384;       // N*256: xr1, later xl2|xr2
  float*    R3    = R2 + n * 256;       // N*256: acc1, later acc2
  float*    R4    = R3 + n * 256;       // N*128: z1, later z2
  float*    R5    = R4 + n * 128;       // N*128: h
  float*    LOGIT = R5 + n * 128;       // EF*2
  float*    DENOM = LOGIT + (size_t)EF * 2;     // N*2
  unsigned* LMAX  = (unsigned*)(DENOM + n * 2); // N*2
  float*    EAS   = (float*)(LMAX + n * 2);     // 32
  float*    SL1   = EAS + 32;           // 256
  float*    SL2   = SL1 + 256;          // 128

  // self-loop edge feature means
  hipMemsetAsync(EAS, 0, 32 * sizeof(float), stream);
  ea_reduce_kernel<<<64, 256, 0, stream>>>(ea, EAS);
  sl_ea_kernel<<<1, 384, 0, stream>>>(EAS, We1, We2, SL1, SL2);

  // ---- GATv2 layer 1 (H=2) ----
  node_gemm_wmma<256, 256, false><<<dim3(NN / 16, 4), 128, 0, stream>>>(x, Wl1, bl1, R1);
  node_gemm_wmma<256, 256, false><<<dim3(NN / 16, 4), 128, 0, stream>>>(x, Wr1, br1, R2);
  hipMemsetAsync(LMAX, 0, n * 2 * sizeof(unsigned), stream);
  hipMemsetAsync(DENOM, 0, n * 2 * sizeof(float), stream);
  gat_logit_kernel<2><<<(EF * 2) / 8, 256, 0, stream>>>(R1, R2, srcp, dstp, ea, SL1, We1, att1,
                                                        LOGIT, LMAX);
  gat_softmax_kernel<2><<<(EF * 2 + 255) / 256, 256, 0, stream>>>(LOGIT, LMAX, DENOM, dstp);
  hipMemsetAsync(R3, 0, n * 256 * sizeof(float), stream);
  gat_aggregate_kernel<2><<<(EF * 2) / 8, 256, 0, stream>>>(R1, LOGIT, DENOM, srcp, dstp, R3);
  finalize_elu_kernel<2><<<(NN * HIDC) / 256, 256, 0, stream>>>(R3, bias1, R4);

  // ---- GATv2 layer 2 (H=1) ----
  float* XL2 = R2;
  float* XR2 = R2 + n * 128;
  node_gemm_wmma<128, 128, false><<<dim3(NN / 16, 2), 128, 0, stream>>>(R4, Wl2, bl2, XL2);
  node_gemm_wmma<128, 128, false><<<dim3(NN / 16, 2), 128, 0, stream>>>(R4, Wr2, br2, XR2);
  hipMemsetAsync(LMAX, 0, n * sizeof(unsigned), stream);
  hipMemsetAsync(DENOM, 0, n * sizeof(float), stream);
  gat_logit_kernel<1><<<EF / 8, 256, 0, stream>>>(XL2, XR2, srcp, dstp, ea, SL2, We2, att2,
                                                  LOGIT, LMAX);
  gat_softmax_kernel<1><<<(EF + 255) / 256, 256, 0, stream>>>(LOGIT, LMAX, DENOM, dstp);
  hipMemsetAsync(R3, 0, n * 128 * sizeof(float), stream);
  gat_aggregate_kernel<1><<<EF / 8, 256, 0, stream>>>(XL2, LOGIT, DENOM, srcp, dstp, R3);
  finalize_elu_kernel<1><<<(NN * HIDC) / 256, 256, 0, stream>>>(R3, bias2, R4);

  // ---- GRU (h_prev = 0): gi = z2 @ W_ih^T + b_ih ----
  node_gemm_wmma<128, 384, true><<<dim3(NN / 16, 6), 128, 0, stream>>>(R4, W_ih, b_ih, R1);
  gru_kernel<<<(NN * HIDC) / 256, 256, 0, stream>>>(R1, b_hh, R5);

  // ---- fused edge classifier ----
  edge_classifier_kernel<<<EE / 16, 256, 0, stream>>>(R5, srcp, dstp, ea, Wc1, bc1, Wc2, bc2,
                                                      Wc3, bc3, outp);
  (void)in_sizes; (void)n_in; (void)out_size; (void)ws_size;
}